// GlassBlur_12275016532594
// MI455X (gfx1250) — compile-verified
//
#include <hip/hip_runtime.h>

#define HH 224
#define WW 224
#define CC 3
#define BB 64
#define NSW (223*223)   // 49729 swaps per image
#define STW 240         // padded stage width (15 tiles of 16)
#define CHUNK 2048      // delta pairs staged per round

typedef float v2f __attribute__((ext_vector_type(2)));
typedef float v8f __attribute__((ext_vector_type(8)));

__device__ __forceinline__ void wait_asynccnt0() {
#if __has_builtin(__builtin_amdgcn_s_wait_asynccnt)
  __builtin_amdgcn_s_wait_asynccnt(0);
#else
  asm volatile("s_wait_asynccnt 0" ::: "memory");
#endif
}

// Blur pass over one image.
// pass==0: source = global x (quantize), sink = packed bytes in LDS (atomicOr)
// pass==1: source = packed bytes in LDS, sink = global out (floor/255)
__device__ __forceinline__ void blur_pass(
    int b, int pass,
    const float* __restrict__ x, float* __restrict__ out,
    unsigned int* imgPacked, float* stage0, float* out1,
    const v2f band[5],
    int tid, int wid, int l16, int hi)
{
  for (int c = 0; c < CC; ++c) {
    for (int s = 0; s < 14; ++s) {
      // ---- phase L: stage 20 rows x 240 cols with edge clamp ----
      for (int i = tid; i < 20 * STW; i += 256) {
        int r  = i / STW;
        int sc = i - r * STW;
        int row = 16 * s + r - 2; row = min(max(row, 0), HH - 1);
        int col = sc - 2;         col = min(max(col, 0), WW - 1);
        float v;
        if (pass == 0) {
          float t = x[((size_t)(b * CC + c) * HH + row) * WW + col] * 255.0f;
          t = fminf(fmaxf(t, 0.0f), 255.0f);
          v = floorf(t);
        } else {
          v = (float)((imgPacked[row * WW + col] >> (8 * c)) & 0xffu);
        }
        stage0[i] = v;
      }
      __syncthreads();

      // ---- phase V: vertical blur, 15 tiles of 16 cols ----
      // D(16x16) = A_band(16x20) x In(20x16), 5 chained K=4 WMMAs
      for (int tile = wid; tile < 15; tile += 8) {
        int n0 = tile * 16;
        v8f acc = {};
#pragma unroll
        for (int t = 0; t < 5; ++t) {
          v2f bb;
          int rr = 4 * t + 2 * hi;                    // B: K = 4t + 2*hi + v
          bb[0] = stage0[rr * STW + n0 + l16];
          bb[1] = stage0[(rr + 1) * STW + n0 + l16];
          acc = __builtin_amdgcn_wmma_f32_16x16x4_f32(
              false, band[t], false, bb, (short)0, acc, false, false);
        }
#pragma unroll
        for (int j = 0; j < 8; ++j)                   // D: M = j + 8*hi, N = l16
          out1[(j + 8 * hi) * STW + n0 + l16] = acc[j];
      }
      __syncthreads();

      // ---- phase H: horizontal blur, 14 tiles of 16 cols ----
      // D(16x16) = In(16x20) x B_band(20x16)
      for (int tile = wid; tile < 14; tile += 8) {
        int n0 = tile * 16;
        v8f acc = {};
#pragma unroll
        for (int t = 0; t < 5; ++t) {
          v2f aa;
          int cbase = n0 + 4 * t + 2 * hi;            // A: M = l16, K = 4t + 2*hi + v
          aa[0] = out1[l16 * STW + cbase];
          aa[1] = out1[l16 * STW + cbase + 1];
          acc = __builtin_amdgcn_wmma_f32_16x16x4_f32(
              false, aa, false, band[t], (short)0, acc, false, false);
        }
        if (pass == 0) {
#pragma unroll
          for (int j = 0; j < 8; ++j) {
            int row = 16 * s + j + 8 * hi;
            int col = n0 + l16;
            float v = floorf(fminf(fmaxf(acc[j], 0.0f), 255.0f));
            atomicOr(&imgPacked[row * WW + col], ((unsigned int)v) << (8 * c));
          }
        } else {
#pragma unroll
          for (int j = 0; j < 8; ++j) {
            int row = 16 * s + j + 8 * hi;
            int col = n0 + l16;
            float v = floorf(fminf(fmaxf(acc[j], 0.0f), 255.0f)) / 255.0f;
            out[((size_t)(b * CC + c) * HH + row) * WW + col] = v;
          }
        }
      }
      __syncthreads();
    }
  }
}

__global__ __launch_bounds__(256) void GlassBlur_kernel(
    const float* __restrict__ x, const int* __restrict__ deltas,
    float* __restrict__ out)
{
  __shared__ unsigned int imgPacked[HH * WW];  // 196 KB: RGB packed per pixel
  __shared__ float stage0[20 * STW];           // 18.75 KB: strip + halo
  __shared__ float out1[16 * STW];             // 15 KB: after vertical pass
  __shared__ int   dbuf[CHUNK * 2];            // 16 KB: staged delta pairs

  const int b    = blockIdx.x;
  const int tid  = threadIdx.x;
  const int lane = tid & 31;
  const int wid  = tid >> 5;
  const int l16  = lane & 15;
  const int hi   = lane >> 4;

  // Gaussian weights, float64 normalized then cast (matches reference)
  const double k1 = 0.04393693362340742;    // exp(-3.125)
  const double k2 = 3.726653172078671e-06;  // exp(-12.5)
  const double S  = 1.0 + 2.0 * k1 + 2.0 * k2;
  float Wk[5];
  Wk[0] = Wk[4] = (float)(k2 / S);
  Wk[1] = Wk[3] = (float)(k1 / S);
  Wk[2] = (float)(1.0 / S);

  // Banded matrix fragment: value(t, v) = w[4t + 2*hi + v - l16], 0 outside band.
  // Serves as A (16x20, A[m][K]=w[K-m]) vertically and B (20x16, B[K][n]=w[K-n])
  // horizontally under the symmetric 32-bit WMMA operand layouts.
  v2f band[5];
#pragma unroll
  for (int t = 0; t < 5; ++t)
#pragma unroll
    for (int v = 0; v < 2; ++v) {
      int d = 4 * t + 2 * hi + v - l16;
      band[t][v] = (d >= 0 && d <= 4) ? Wk[d] : 0.0f;
    }

  for (int i = tid; i < HH * WW; i += 256) imgPacked[i] = 0u;
  __syncthreads();

  // Stage 1: quantize + blur1 -> packed u8x3 image in LDS
  blur_pass(b, 0, x, out, imgPacked, stage0, out1, band, tid, wid, l16, hi);
  __syncthreads();

  // Stage 2: 49729 strictly sequential neighbor swaps on the LDS-resident image.
  // Delta pairs are DMA'd HBM->LDS with the gfx1250 async-to-LDS engine
  // (no VGPR round trip, tracked by ASYNCcnt); the dependent swap chain then
  // runs at LDS latency on lane 0.
  const int* db = deltas + (size_t)b * NSW * 2;
  unsigned dbuf_base =
      (unsigned)(uintptr_t)(__attribute__((address_space(3))) int*)dbuf;
  for (int base = 0; base < NSW; base += CHUNK) {
    int cnt = min(CHUNK, NSW - base);
    for (int j = tid; j < cnt; j += 256) {
      unsigned ldsoff = dbuf_base + (unsigned)j * 8u;   // one int2 pair
      const int* g = &db[2 * (base + j)];
      asm volatile("global_load_async_to_lds_b64 %0, %1, off"
                   :: "v"(ldsoff), "v"(g) : "memory");
    }
    wait_asynccnt0();
    __syncthreads();
    if (tid == 0) {
      for (int j = 0; j < cnt; ++j) {
        int i  = base + j;
        int q  = i / 223;
        int y  = 223 - q;
        int xx = 223 - (i - q * 223);
        int dy = dbuf[2 * j] - 1;
        int dx = dbuf[2 * j + 1] - 1;
        int yp = min(max(y + dy, 0), HH - 1);
        int xp = min(max(xx + dx, 0), WW - 1);
        int ia = y * WW + xx;
        int ib = yp * WW + xp;
        unsigned int a  = imgPacked[ia];
        unsigned int bb = imgPacked[ib];
        imgPacked[ia] = bb;   // set order matches reference (self-swap = identity)
        imgPacked[ib] = a;
      }
    }
    __syncthreads();
  }

  // Stage 3: blur2 -> floor/255 -> global out
  blur_pass(b, 1, x, out, imgPacked, stage0, out1, band, tid, wid, l16, hi);
}

extern "C" void kernel_launch(void* const* d_in, const int* in_sizes, int n_in,
                              void* d_out, int out_size, void* d_ws, size_t ws_size,
                              hipStream_t stream) {
  const float* x      = (const float*)d_in[0];
  const int*   deltas = (const int*)d_in[1];
  float*       out    = (float*)d_out;
  (void)in_sizes; (void)n_in; (void)out_size; (void)d_ws; (void)ws_size;
  GlassBlur_kernel<<<dim3(BB), dim3(256), 0, stream>>>(x, deltas, out);
}